// TopKRouter_64673617543277
// MI455X (gfx1250) — compile-verified
//
#include <hip/hip_runtime.h>
#include <hip/hip_bf16.h>
#include <math.h>

typedef float v2f __attribute__((ext_vector_type(2)));
typedef float v8f __attribute__((ext_vector_type(8)));

#define HIDDEN 2048
#define NTOK   32768
#define NEXP   8
#define WAVES_PER_BLOCK 8
#define TOK_PER_WAVE    16
#define Z_LOSS_COEF  0.001f
#define LB_LOSS_COEF 0.01f

// ---------------------------------------------------------------------------
// ws layout (floats): [0..7]=importance, [8..15]=load, [16]=sum(lse^2)
// ---------------------------------------------------------------------------
__global__ __launch_bounds__(32) void router_init(float* __restrict__ acc) {
    if (threadIdx.x < 17) acc[threadIdx.x] = 0.0f;
}

// ---------------------------------------------------------------------------
// Main pass: logits via V_WMMA_F32_16X16X4_F32 (f32 in/out — exact, and the
// kernel is HBM-bound on streaming x anyway). One wave owns a 16-token tile.
// B columns 8..15 of the 16-wide expert dim are DON'T-CARE: we clamp the
// expert row index (row & 7) instead of zero-padding, so every load in the
// hot loop is unconditional (no exec-mask branches). Only experts 0..7 are
// read back from the C tile.
// ---------------------------------------------------------------------------
__global__ __launch_bounds__(WAVES_PER_BLOCK * 32) void router_main(
    const float* __restrict__ x,        // [NTOK][HIDDEN]
    const float* __restrict__ W,        // [NEXP][HIDDEN]
    float* __restrict__ out_experts,    // [NTOK][2] (indices as float)
    float* __restrict__ out_scores,     // [NTOK][2]
    float* __restrict__ acc)            // global accumulators (17 floats)
{
    __shared__ float smem[WAVES_PER_BLOCK][TOK_PER_WAVE][16];
    __shared__ float s_imp[NEXP];
    __shared__ float s_load[NEXP];
    __shared__ float s_z;

    const int lane = threadIdx.x & 31;
    const int wave = threadIdx.x >> 5;

    if (threadIdx.x < NEXP) { s_imp[threadIdx.x] = 0.0f; s_load[threadIdx.x] = 0.0f; }
    if (threadIdx.x == 0)   { s_z = 0.0f; }
    __syncthreads();

    const int tile  = blockIdx.x * WAVES_PER_BLOCK + wave;   // 16 tokens per tile
    const int row   = lane & 15;          // A: token row in tile; B: expert column
    const int khalf = (lane >> 4) * 2;    // lanes 0-15 -> K 0,1 ; lanes 16-31 -> K 2,3

    const float* __restrict__ arow = x + (size_t)(tile * TOK_PER_WAVE + row) * HIDDEN + khalf;
    const float* __restrict__ brow = W + (size_t)(row & 7) * HIDDEN + khalf;  // always valid

    v8f c = {0.f, 0.f, 0.f, 0.f, 0.f, 0.f, 0.f, 0.f};

    for (int k = 0; k < HIDDEN; k += 16) {
        __builtin_prefetch(arow + k + 512, 0, 3);   // global_prefetch_b8, ~2KB ahead
        v2f a0 = *(const v2f*)(arow + k);
        v2f a1 = *(const v2f*)(arow + k + 4);
        v2f a2 = *(const v2f*)(arow + k + 8);
        v2f a3 = *(const v2f*)(arow + k + 12);
        v2f b0 = *(const v2f*)(brow + k);
        v2f b1 = *(const v2f*)(brow + k + 4);
        v2f b2 = *(const v2f*)(brow + k + 8);
        v2f b3 = *(const v2f*)(brow + k + 12);
        // (neg_a, A, neg_b, B, c_mod, C, reuse_a, reuse_b)
        c = __builtin_amdgcn_wmma_f32_16x16x4_f32(false, a0, false, b0, (short)0, c, false, false);
        c = __builtin_amdgcn_wmma_f32_16x16x4_f32(false, a1, false, b1, (short)0, c, false, false);
        c = __builtin_amdgcn_wmma_f32_16x16x4_f32(false, a2, false, b2, (short)0, c, false, false);
        c = __builtin_amdgcn_wmma_f32_16x16x4_f32(false, a3, false, b3, (short)0, c, false, false);
    }

    // C layout: VGPR r -> token (lane<16 ? r : 8+r), expert = lane&15.
    // Transpose through LDS so each of lanes 0..15 owns one token's 8 logits.
    const int tbase = (lane >> 4) * 8;
    #pragma unroll
    for (int r = 0; r < 8; ++r)
        smem[wave][tbase + r][row] = c[r];
    __syncthreads();

    if (lane < TOK_PER_WAVE) {
        const int t = tile * TOK_PER_WAVE + lane;
        float l[NEXP];
        #pragma unroll
        for (int e = 0; e < NEXP; ++e) l[e] = smem[wave][lane][e];

        float m = l[0];
        #pragma unroll
        for (int e = 1; e < NEXP; ++e) m = fmaxf(m, l[e]);

        float p[NEXP];
        float s = 0.0f;
        #pragma unroll
        for (int e = 0; e < NEXP; ++e) { p[e] = __expf(l[e] - m); s += p[e]; }
        const float inv = 1.0f / s;
        #pragma unroll
        for (int e = 0; e < NEXP; ++e) p[e] *= inv;
        const float lse = m + __logf(s);

        // top-2, first occurrence wins ties (matches jax.lax.top_k)
        int e1 = 0; float s1 = p[0];
        #pragma unroll
        for (int e = 1; e < NEXP; ++e) if (p[e] > s1) { s1 = p[e]; e1 = e; }
        int e2 = (e1 == 0) ? 1 : 0; float s2 = p[e2];
        #pragma unroll
        for (int e = 0; e < NEXP; ++e) if (e != e1 && p[e] > s2) { s2 = p[e]; e2 = e; }

        const float denom = fmaxf(s1 + s2, 1e-9f);
        out_experts[2 * t]     = (float)e1;
        out_experts[2 * t + 1] = (float)e2;
        out_scores[2 * t]      = s1 / denom;
        out_scores[2 * t + 1]  = s2 / denom;

        atomicAdd(&s_z, lse * lse);
        atomicAdd(&s_load[e1], 1.0f);
        #pragma unroll
        for (int e = 0; e < NEXP; ++e) atomicAdd(&s_imp[e], p[e]);
    }
    __syncthreads();

    if (threadIdx.x < NEXP) {
        atomicAdd(&acc[threadIdx.x],        s_imp[threadIdx.x]);
        atomicAdd(&acc[NEXP + threadIdx.x], s_load[threadIdx.x]);
    }
    if (threadIdx.x == 0) atomicAdd(&acc[16], s_z);
}

// ---------------------------------------------------------------------------
__global__ __launch_bounds__(32) void router_finalize(const float* __restrict__ acc,
                                                      float* __restrict__ out_tail) {
    if (threadIdx.x == 0) {
        float imp_sum = 0.0f, load_sum = 0.0f;
        #pragma unroll
        for (int i = 0; i < NEXP; ++i) { imp_sum += acc[i]; load_sum += acc[NEXP + i]; }
        imp_sum  = fmaxf(imp_sum,  1e-9f);
        load_sum = fmaxf(load_sum, 1e-9f);
        float lb = 0.0f;
        #pragma unroll
        for (int i = 0; i < NEXP; ++i)
            lb += (acc[i] / imp_sum) * (acc[NEXP + i] / load_sum);
        out_tail[0] = (acc[16] / (float)NTOK) * Z_LOSS_COEF;                 // z_loss
        out_tail[1] = lb * (float)(NEXP * NEXP) * LB_LOSS_COEF;              // lb_loss
    }
}

// ---------------------------------------------------------------------------
extern "C" void kernel_launch(void* const* d_in, const int* in_sizes, int n_in,
                              void* d_out, int out_size, void* d_ws, size_t ws_size,
                              hipStream_t stream) {
    const float* x = (const float*)d_in[0];   // [32768, 2048] f32
    const float* W = (const float*)d_in[1];   // [8, 2048] f32
    float* out = (float*)d_out;               // experts(65536) | scores(65536) | z | lb
    float* acc = (float*)d_ws;

    router_init<<<1, 32, 0, stream>>>(acc);

    const int tiles  = NTOK / TOK_PER_WAVE;           // 2048 waves
    const int blocks = tiles / WAVES_PER_BLOCK;       // 256 blocks of 256 threads
    router_main<<<blocks, WAVES_PER_BLOCK * 32, 0, stream>>>(
        x, W, out, out + 2 * NTOK, acc);

    router_finalize<<<1, 32, 0, stream>>>(acc, out + 4 * NTOK);
}